// Feature_Correlation_Matching_29953101922623
// MI455X (gfx1250) — compile-verified
//
#include <hip/hip_runtime.h>
#include <math.h>

// CDNA5 fp32 WMMA fragment types
typedef __attribute__((ext_vector_type(2))) float v2f;
typedef __attribute__((ext_vector_type(8))) float v8f;

#define BATCH 4
#define N 4096
#define C 512
#define TL_THRESH 0.6f
#define SCALE_L 2.0f
#define EPS_D 1e-12f

// 512 + 4 floats padding: fragment-read bank = (4*row + k) % 64 -> all 64 banks,
// zero conflicts for the b64 fragment loads.
#define LDS_STRIDE 516

#define ROWS_PER_BLOCK 128   // 4 waves * 32 rows each
#define JT_PER_HALF    128   // 256 column tiles split across 2 blocks

// ---------------- Kernel 1: row squared norms (one wave per row) ----------------
__global__ void row_sqnorm_kernel(const float* __restrict__ x, float* __restrict__ sq) {
    const int row  = blockIdx.x * (blockDim.x >> 5) + (threadIdx.x >> 5); // [0, BATCH*N)
    const int lane = threadIdx.x & 31;
    const float* p = x + (size_t)row * C;
    float s = 0.0f;
    #pragma unroll
    for (int i = 0; i < C; i += 32 * 4) {
        float4 v = *(const float4*)(p + i + lane * 4);
        s += v.x * v.x + v.y * v.y + v.z * v.z + v.w * v.w;
    }
    #pragma unroll
    for (int m = 16; m >= 1; m >>= 1) s += __shfl_xor(s, m, 32);
    if (lane == 0) sq[row] = s;
}

// Branchless insert of x into sorted running minima m1 <= m2 <= m3.
__device__ __forceinline__ void insert3(float& m1, float& m2, float& m3, float x) {
    float a1 = fmaxf(m1, x);  m1 = fminf(m1, x);
    float a2 = fmaxf(m2, a1); m2 = fminf(m2, a1);
    m3 = fminf(m3, a2);
}

__device__ __forceinline__ v8f wmma_f32(v2f a, v2f b, v8f c) {
    return __builtin_amdgcn_wmma_f32_16x16x4_f32(false, a, false, b, (short)0, c,
                                                 false, false);
}

// ---------------- Kernel 2: fused Gram (WMMA f32) + partial kNN stats -----------
// Grid: 4 batches * 32 row-blocks * 2 column-halves = 256 blocks.
// Block: 128 threads = 4 waves; 128 rows; each wave owns 32 rows (two 16-row tiles).
// Next B panel (contiguous 32 KB) is prefetched with global_prefetch during the
// K-loop so the synchronous LDS staging hits WGP$/L2 -- zero register cost.
__global__ void __launch_bounds__(128, 1)
knn_stats_kernel(const float* __restrict__ x, const float* __restrict__ sq,
                 float4* __restrict__ part) {
    extern __shared__ float lds[];
    float* Alds = lds;                                 // 128 * LDS_STRIDE
    float* Blds = lds + ROWS_PER_BLOCK * LDS_STRIDE;   // 16 * LDS_STRIDE

    const int bb    = blockIdx.x >> 6;       // batch
    const int rem   = blockIdx.x & 63;
    const int jhalf = rem >> 5;              // which half of the column range
    const int rb    = rem & 31;              // row-block (128 rows each)
    const int row0  = rb * ROWS_PER_BLOCK;
    const int jt0   = jhalf * JT_PER_HALF;
    const float* xb  = x  + (size_t)bb * N * C;
    const float* sqb = sq + bb * N;

    const int tid   = threadIdx.x;           // also the float4 column inside a row
    const int wave  = tid >> 5;
    const int lane  = tid & 31;
    const int lhalf = lane >> 4;             // selects K-pair / M-half
    const int l16   = lane & 15;

    // ---- stage A panel: 128 rows x 512 f32 -> LDS (coalesced float4) ----
    #pragma unroll 4
    for (int r = 0; r < ROWS_PER_BLOCK; ++r) {
        float4 v = *(const float4*)(xb + (size_t)(row0 + r) * C + tid * 4);
        *(float4*)(Alds + r * LDS_STRIDE + tid * 4) = v;
    }

    const int base = row0 + wave * 32;       // this wave's first row
    float sqr0[8], sqr1[8];
    #pragma unroll
    for (int v = 0; v < 8; ++v) {
        sqr0[v] = sqb[base + v + 8 * lhalf];          // tile 0: rows base..base+15
        sqr1[v] = sqb[base + 16 + v + 8 * lhalf];     // tile 1: rows base+16..base+31
    }

    // Running stats on squared distance (sqrt deferred; monotone).
    float m1a[8], m2a[8], m3a[8], mxa[8];
    float m1b[8], m2b[8], m3b[8], mxb[8];
    #pragma unroll
    for (int v = 0; v < 8; ++v) {
        m1a[v] = m2a[v] = m3a[v] = 3.0e38f; mxa[v] = 0.0f;
        m1b[v] = m2b[v] = m3b[v] = 3.0e38f; mxb[v] = 0.0f;
    }

    const float* Abase0 = Alds + (wave * 32 + l16) * LDS_STRIDE + 2 * lhalf;
    const float* Abase1 = Abase0 + 16 * LDS_STRIDE;
    const float* Bbase  = Blds + l16 * LDS_STRIDE + 2 * lhalf;

    for (int t = 0; t < JT_PER_HALF; ++t) {
        const int jt = jt0 + t;
        __syncthreads();                     // previous B panel fully consumed
        // ---- stage B panel: 16 feature rows x 512 f32 -> LDS (hits cache) ----
        #pragma unroll
        for (int it = 0; it < 16; ++it) {
            float4 v = *(const float4*)(xb + (size_t)(jt * 16 + it) * C + tid * 4);
            *(float4*)(Blds + it * LDS_STRIDE + tid * 4) = v;
        }
        __syncthreads();

        // ---- prefetch next B panel: contiguous 32 KB, one touch per 128B line ----
        {
            const int jn = jt0 + ((t + 1) & (JT_PER_HALF - 1)); // wrap: harmless
            const char* nb = (const char*)(xb + (size_t)jn * 16 * C);
            __builtin_prefetch(nb + tid * 256, 0, 3);
            __builtin_prefetch(nb + tid * 256 + 128, 0, 3);
        }

        // ---- two 16x16 Gram tiles, 4 independent WMMA chains over K=512 ----
        v8f acc0a = {}, acc0b = {}, acc1a = {}, acc1b = {};
        #pragma unroll 4
        for (int k = 0; k < C; k += 8) {
            v2f b0 = *(const v2f*)(Bbase + k);
            v2f p0 = *(const v2f*)(Abase0 + k);
            v2f q0 = *(const v2f*)(Abase1 + k);
            acc0a = wmma_f32(p0, b0, acc0a);
            acc1a = wmma_f32(q0, b0, acc1a);
            v2f b1 = *(const v2f*)(Bbase + k + 4);
            v2f p1 = *(const v2f*)(Abase0 + k + 4);
            v2f q1 = *(const v2f*)(Abase1 + k + 4);
            acc0b = wmma_f32(p1, b1, acc0b);
            acc1b = wmma_f32(q1, b1, acc1b);
        }

        // ---- fold both tiles into running per-row stats ----
        const float sqc = sqb[jt * 16 + l16];          // column norm (N = l16)
        #pragma unroll
        for (int v = 0; v < 8; ++v) {
            float g0 = acc0a[v] + acc0b[v];
            float g1 = acc1a[v] + acc1b[v];
            float d0 = fmaxf(sqr0[v] + sqc - 2.0f * g0, 0.0f);
            float d1 = fmaxf(sqr1[v] + sqc - 2.0f * g1, 0.0f);
            mxa[v] = fmaxf(mxa[v], d0);
            mxb[v] = fmaxf(mxb[v], d1);
            insert3(m1a[v], m2a[v], m3a[v], d0);
            insert3(m1b[v], m2b[v], m3b[v], d1);
        }
    }

    // ---- cross-lane merge over the 16 lanes sharing each row ----
    #pragma unroll
    for (int msk = 1; msk <= 8; msk <<= 1) {
        #pragma unroll
        for (int v = 0; v < 8; ++v) {
            float o1 = __shfl_xor(m1a[v], msk, 32);
            float o2 = __shfl_xor(m2a[v], msk, 32);
            float o3 = __shfl_xor(m3a[v], msk, 32);
            float om = __shfl_xor(mxa[v], msk, 32);
            insert3(m1a[v], m2a[v], m3a[v], o1);
            insert3(m1a[v], m2a[v], m3a[v], o2);
            insert3(m1a[v], m2a[v], m3a[v], o3);
            mxa[v] = fmaxf(mxa[v], om);
            o1 = __shfl_xor(m1b[v], msk, 32);
            o2 = __shfl_xor(m2b[v], msk, 32);
            o3 = __shfl_xor(m3b[v], msk, 32);
            om = __shfl_xor(mxb[v], msk, 32);
            insert3(m1b[v], m2b[v], m3b[v], o1);
            insert3(m1b[v], m2b[v], m3b[v], o2);
            insert3(m1b[v], m2b[v], m3b[v], o3);
            mxb[v] = fmaxf(mxb[v], om);
        }
    }

    // ---- write partial stats (one float4 per row per column-half) ----
    if (l16 == 0) {
        #pragma unroll
        for (int v = 0; v < 8; ++v) {
            int r0 = base + v + 8 * lhalf;
            int r1 = base + 16 + v + 8 * lhalf;
            part[(size_t)(bb * N + r0) * 2 + jhalf] =
                make_float4(m1a[v], m2a[v], m3a[v], mxa[v]);
            part[(size_t)(bb * N + r1) * 2 + jhalf] =
                make_float4(m1b[v], m2b[v], m3b[v], mxb[v]);
        }
    }
}

// ---------------- Kernel 3: merge the two column-halves + prediction ------------
__global__ void merge_pred_kernel(const float4* __restrict__ part,
                                  float* __restrict__ out) {
    int i = blockIdx.x * blockDim.x + threadIdx.x;    // [0, BATCH*N)
    float4 p0 = part[(size_t)i * 2 + 0];
    float4 p1 = part[(size_t)i * 2 + 1];
    float m1 = p0.x, m2 = p0.y, m3 = p0.z, mx = fmaxf(p0.w, p1.w);
    insert3(m1, m2, m3, p1.x);
    insert3(m1, m2, m3, p1.y);
    insert3(m1, m2, m3, p1.z);
    float d1 = sqrtf(m2 + EPS_D);   // 2nd smallest distance (1st is self ~ 0)
    float dn = sqrtf(m3 + EPS_D);   // 3rd smallest
    float dm = sqrtf(mx + EPS_D);   // max
    out[i] = (d1 / dn < TL_THRESH) ? 2.0f / (1.0f + expf(d1))
                                   : 2.0f / (1.0f + SCALE_L * expf(dm));
}

extern "C" void kernel_launch(void* const* d_in, const int* in_sizes, int n_in,
                              void* d_out, int out_size, void* d_ws, size_t ws_size,
                              hipStream_t stream) {
    const float* x = (const float*)d_in[0];      // (4, 64, 64, 512) f32
    float* out = (float*)d_out;                  // (4, 64, 64) f32 = 16384
    float* sq  = (float*)d_ws;                   // BATCH*N f32 = 64 KB
    float4* part = (float4*)(sq + BATCH * N);    // BATCH*N*2 float4 = 512 KB

    // Kernel 1: 16384 rows, 8 waves/block -> 2048 blocks
    row_sqnorm_kernel<<<(BATCH * N) / 8, 256, 0, stream>>>(x, sq);

    // Kernel 2: 256 blocks (4 batches x 32 row-blocks x 2 column-halves).
    size_t lds_bytes = (size_t)(ROWS_PER_BLOCK + 16) * LDS_STRIDE * sizeof(float);
    knn_stats_kernel<<<BATCH * 32 * 2, 128, lds_bytes, stream>>>(x, sq, part);

    // Kernel 3: merge halves + sigmoid prediction.
    merge_pred_kernel<<<(BATCH * N) / 256, 256, 0, stream>>>(part, out);
}